// WinCorrTorch_VFA_81647328297401
// MI455X (gfx1250) — compile-verified
//
#include <hip/hip_runtime.h>

typedef __attribute__((ext_vector_type(2))) float v2f;
typedef __attribute__((ext_vector_type(8))) float v8f;

#define VOL   (96u * 96u * 96u)   /* 884736 voxels per channel */
#define PLANE (96u * 96u)         /* 9216 */

__device__ __forceinline__ int clamp95(int v) { return v < 0 ? 0 : (v > 95 ? 95 : v); }

// One wave handles one (z, y, x-tile-of-14) task.
// For each of the 9 (dz,dy) pairs it computes a 16x16x64 fp32 gram via
// V_WMMA_F32_16X16X4_F32 and extracts the three dx diagonals.
//
// All channel-block offsets (kb*4*VOL) are kept uniform so loads lower to
// GVS form: global_load_b32 v, v_off, s[base] [offset:imm24]. Only the
// spatial offset lives in a VGPR and it is shared by all 32 loads of a gram.
__global__ __launch_bounds__(256)
void wincorr_wmma_kernel(const float* __restrict__ fixedf,
                         const float* __restrict__ moving,
                         float* __restrict__ out)
{
    __shared__ float lds[8][16 * 17];   // per-wave D spill tile, padded pitch

    const int lane  = threadIdx.x & 31;
    const int wave  = threadIdx.x >> 5;
    const int n     = lane & 15;        // fragment row (A) / col (B) index
    const int hi    = lane >> 4;        // half-wave selector
    const int khalf = hi * 2;           // K sub-offset {0,2} within each K=4 block

    const int t = blockIdx.x;               // x tile 0..6 (stride 14, outputs 14t..14t+13)
    const int y = blockIdx.y * 8 + wave;    // 0..95
    const int z = blockIdx.z;               // 0..95

    const int xb = 14 * t - 1;              // fragment position of row/col 0
    const int xm = clamp95(xb + n);         // clamped x for both A rows and B cols

    // ---- preload A fragment: A[m,k] = fixed[c, z, y, xb+m], reused by all 9 grams ----
    const unsigned voffA = (unsigned)z * PLANE + (unsigned)y * 96u + (unsigned)xm
                         + (unsigned)khalf * VOL;
    v2f a[16];
#pragma unroll
    for (int kb = 0; kb < 16; ++kb) {
        const float* a0 = fixedf + (size_t)(4 * kb) * VOL;   // uniform -> SGPR base
        a[kb].x = a0[voffA];
        a[kb].y = (a0 + VOL)[voffA];                         // +VOL*4 fits imm24
    }

    float* Dw = lds[wave];

    const int  mo = lane + 1;               // output row handled by this lane (lanes 0..13)
    const int  xo = xb + mo;                // = 14*t + lane
    const bool store_ok = (lane < 14) && (xo < 96);
    const size_t outSpatial = (size_t)z * PLANE + (size_t)y * 96u + (size_t)(store_ok ? xo : 0);

    for (int dz = -1; dz <= 1; ++dz) {
        const int zc = clamp95(z + dz);
        for (int dy = -1; dy <= 1; ++dy) {
            const int yc = clamp95(y + dy);

            // ---- load B fragment: B[k,n] = moving[c, zc, yc, xb+n] (edge-padded) ----
            const unsigned voffB = (unsigned)zc * PLANE + (unsigned)yc * 96u + (unsigned)xm
                                 + (unsigned)khalf * VOL;
            v2f b[16];
#pragma unroll
            for (int kb = 0; kb < 16; ++kb) {
                const float* b0 = moving + (size_t)(4 * kb) * VOL;   // uniform -> SGPR base
                b[kb].x = b0[voffB];
                b[kb].y = (b0 + VOL)[voffB];
            }

            // ---- 16x16x64 fp32 gram: 16 chained K=4 WMMAs ----
            v8f acc = {0.f, 0.f, 0.f, 0.f, 0.f, 0.f, 0.f, 0.f};
#pragma unroll
            for (int kb = 0; kb < 16; ++kb) {
                acc = __builtin_amdgcn_wmma_f32_16x16x4_f32(
                        /*neg_a=*/false, a[kb],
                        /*neg_b=*/false, b[kb],
                        /*c_mod=*/(short)0, acc,
                        /*reuse_a=*/false, /*reuse_b=*/false);
            }

            // ---- spill D tile to LDS: D[M,N] -> Dw[M*17 + N] ----
#pragma unroll
            for (int j = 0; j < 8; ++j)
                Dw[(j + hi * 8) * 17 + n] = acc[j];
            asm volatile("" ::: "memory");   // keep LDS reads after writes (same-wave DS is in-order)

            // ---- extract diagonals n = m+dx, scale by 64^{-1/2} = 0.125, store ----
            if (store_ok) {
                const int obase = ((dz + 1) * 3 + (dy + 1)) * 3;
#pragma unroll
                for (int dx = -1; dx <= 1; ++dx) {
                    const float v = Dw[mo * 17 + (mo + dx)] * 0.125f;
                    out[(size_t)(obase + dx + 1) * VOL + outSpatial] = v;
                }
            }
            asm volatile("" ::: "memory");   // reads complete before next gram's LDS writes
        }
    }
}

extern "C" void kernel_launch(void* const* d_in, const int* in_sizes, int n_in,
                              void* d_out, int out_size, void* d_ws, size_t ws_size,
                              hipStream_t stream) {
    const float* fixedf = (const float*)d_in[0];
    const float* moving = (const float*)d_in[1];
    float* out = (float*)d_out;

    dim3 grid(7, 12, 96);    // x tiles (stride 14) x (96/8 y-groups) x z
    dim3 block(256, 1, 1);   // 8 waves; one (z,y,t) task per wave
    hipLaunchKernelGGL(wincorr_wmma_kernel, grid, block, 0, stream,
                       fixedf, moving, out);
}